// GradeWiseLinear_34892314313418
// MI455X (gfx1250) — compile-verified
//
#include <hip/hip_runtime.h>

typedef __attribute__((ext_vector_type(2))) float v2f;
typedef __attribute__((ext_vector_type(8))) float v8f;

// Block-diagonal 16x16 weight: out[o] = sum_i x[i] * M[o][i],
// M[o][i] = w_g[o-lo][i-lo] when o and i fall in the same grade slice, else 0.
// Grade slices: [0,1) d=1, [1,5) d=4, [5,11) d=6, [11,15) d=4, [15,16) d=1.
__device__ __forceinline__ float bd_weight(int o, int i,
    const float* __restrict__ w0, const float* __restrict__ w1,
    const float* __restrict__ w2, const float* __restrict__ w3,
    const float* __restrict__ w4) {
  if (o < 1)       return (i < 1)             ? w0[0]                       : 0.0f;
  else if (o < 5)  return (i >= 1 && i < 5)   ? w1[(o - 1) * 4 + (i - 1)]   : 0.0f;
  else if (o < 11) return (i >= 5 && i < 11)  ? w2[(o - 5) * 6 + (i - 5)]   : 0.0f;
  else if (o < 15) return (i >= 11 && i < 15) ? w3[(o - 11) * 4 + (i - 11)] : 0.0f;
  else             return (i == 15)           ? w4[0]                       : 0.0f;
}

__device__ __forceinline__ float bd_bias(int o,
    const float* __restrict__ b0, const float* __restrict__ b1,
    const float* __restrict__ b2, const float* __restrict__ b3,
    const float* __restrict__ b4) {
  if (o < 1)       return b0[0];
  else if (o < 5)  return b1[o - 1];
  else if (o < 11) return b2[o - 5];
  else if (o < 15) return b3[o - 11];
  else             return b4[0];
}

// One wave handles one 16-row x 16-col fp32 tile per loop iteration:
//   D(16x16) = sum_{c=0..3} A_c(16x4) * B_c(4x16)   via v_wmma_f32_16x16x4_f32
// then D += bias (broadcast along rows), stored back.
__global__ void __launch_bounds__(256) grade_linear_wmma_kernel(
    const float* __restrict__ x, float* __restrict__ out, long long ntiles,
    const float* __restrict__ w0, const float* __restrict__ b0,
    const float* __restrict__ w1, const float* __restrict__ b1,
    const float* __restrict__ w2, const float* __restrict__ b2,
    const float* __restrict__ w3, const float* __restrict__ b3,
    const float* __restrict__ w4, const float* __restrict__ b4) {
  const int lane   = (int)(threadIdx.x & 31);
  const int n      = lane & 15;            // N column (B/C/D) and M row (A)
  const int khalf  = (lane >> 4) << 1;     // 0 for lanes 0-15, 2 for lanes 16-31
  const int mbase  = (lane >> 4) << 3;     // C/D: 0 or 8

  // ---- Build B operand (weight matrix) once per wave. ----
  // B_c layout (4x16 KxN, f32, 2 VGPRs, mirroring the A 16x4 layout):
  //   lanes 0-15 : N=lane,    elem0 = K=4c+0, elem1 = K=4c+1
  //   lanes 16-31: N=lane-16, elem0 = K=4c+2, elem1 = K=4c+3
  v2f bmat[4];
#pragma unroll
  for (int c = 0; c < 4; ++c) {
    const int i0 = c * 4 + khalf;
    bmat[c].x = bd_weight(n, i0,     w0, w1, w2, w3, w4);
    bmat[c].y = bd_weight(n, i0 + 1, w0, w1, w2, w3, w4);
  }
  const float bias = bd_bias(n, b0, b1, b2, b3, b4);

  const long long nwaves = (long long)gridDim.x * (blockDim.x >> 5);
  long long tile = (long long)blockIdx.x * (blockDim.x >> 5) + (threadIdx.x >> 5);

  for (; tile < ntiles; tile += nwaves) {
    const float* __restrict__ xt = x + (tile << 8);   // 16 rows * 16 floats
    float* __restrict__ ot       = out + (tile << 8);

    // Prefetch next tile this wave will process (1 KB; 32 lanes x 32B spacing).
    if (tile + nwaves < ntiles) {
      __builtin_prefetch(xt + (nwaves << 8) + (lane << 3), 0, 0);
    }

    // ---- Load A chunks per ISA 32-bit A 16x4 layout. ----
    //   lanes 0-15 : M=lane,    K = 4c+{0,1}
    //   lanes 16-31: M=lane-16, K = 4c+{2,3}
    v2f a[4];
#pragma unroll
    for (int c = 0; c < 4; ++c) {
      a[c] = *reinterpret_cast<const v2f*>(xt + n * 16 + c * 4 + khalf);
    }

    // ---- Chained fp32 WMMA accumulation. ----
    v8f acc = {};
#pragma unroll
    for (int c = 0; c < 4; ++c) {
      acc = __builtin_amdgcn_wmma_f32_16x16x4_f32(
          /*neg_a=*/false, a[c], /*neg_b=*/false, bmat[c],
          /*c_mod=*/(short)0, acc, /*reuse_a=*/false, /*reuse_b=*/false);
    }

    // ---- Bias + store per 32-bit C/D 16x16 layout. ----
    // VGPR v: lanes 0-15 -> row v, col lane; lanes 16-31 -> row v+8, col lane-16.
#pragma unroll
    for (int v = 0; v < 8; ++v) {
      ot[(mbase + v) * 16 + n] = acc[v] + bias;
    }
  }
}

extern "C" void kernel_launch(void* const* d_in, const int* in_sizes, int n_in,
                              void* d_out, int out_size, void* d_ws, size_t ws_size,
                              hipStream_t stream) {
  (void)n_in; (void)d_ws; (void)ws_size; (void)out_size;
  const float* x  = (const float*)d_in[0];
  const float* w0 = (const float*)d_in[1];
  const float* b0 = (const float*)d_in[2];
  const float* w1 = (const float*)d_in[3];
  const float* b1 = (const float*)d_in[4];
  const float* w2 = (const float*)d_in[5];
  const float* b2 = (const float*)d_in[6];
  const float* w3 = (const float*)d_in[7];
  const float* b3 = (const float*)d_in[8];
  const float* w4 = (const float*)d_in[9];
  const float* b4 = (const float*)d_in[10];

  const long long nelem  = (long long)in_sizes[0];   // 8*2048*128*16
  const long long ntiles = nelem >> 8;               // 256 fp32 per 16x16 tile

  const int block = 256;                             // 8 waves
  long long waves_needed = ntiles;                   // 1 tile/wave minimum
  long long blocks = (waves_needed + 7) / 8;
  if (blocks > 4096) blocks = 4096;                  // grid-stride covers the rest
  if (blocks < 1) blocks = 1;

  grade_linear_wmma_kernel<<<(dim3)(unsigned)blocks, block, 0, stream>>>(
      x, (float*)d_out, ntiles, w0, b0, w1, b1, w2, b2, w3, b3, w4, b4);
}